// DecoderCell_58841051955875
// MI455X (gfx1250) — compile-verified
//
#include <hip/hip_runtime.h>

typedef __attribute__((ext_vector_type(2))) float v2f;
typedef __attribute__((ext_vector_type(8))) float v8f;

#define NEG_INF (-1e9f)

#define N_NODES 1000
#define NP      1024        // padded node count (8 tiles * 128)
#define DIM     128
#define TSTRIDE 132         // padded LDS tile row stride (floats) to spread banks

// Issue one 128x128 f32 tile of X (rows clamped at node 999) into an LDS tile
// buffer via async-to-LDS b128 transfers. 256 threads * 16 issues * 16B = 64KB.
// Tracked on ASYNCcnt; caller must s_wait_asynccnt + barrier before consuming.
__device__ __forceinline__ void async_tile_load(const float* __restrict__ Xb,
                                                float* tileBuf, int nodeBase,
                                                int tid) {
#pragma unroll
    for (int i = 0; i < 16; ++i) {
        const int e   = (i * 256 + tid) * 4;   // flat float index, 16B granules
        const int row = e >> 7;                // 0..127
        const int col = e & 127;
        int node = nodeBase + row;
        node = node < N_NODES ? node : (N_NODES - 1);
        const float* gptr = Xb + (size_t)node * DIM + col;
        const unsigned ldsOff =
            (unsigned)(uintptr_t)(tileBuf + row * TSTRIDE + col);
        asm volatile("global_load_async_to_lds_b128 %0, %1, off"
                     :: "v"(ldsOff), "v"(gptr) : "memory");
    }
}

__device__ __forceinline__ void wait_async0() {
    asm volatile("s_wait_asynccnt 0" ::: "memory");
}

__launch_bounds__(256)
__global__ void decoder_cell_kernel(
    const float* __restrict__ X,            // [B,1000,128]
    const float* __restrict__ Gemb,         // [B,128]
    const float* __restrict__ Sctx,         // [B,130]
    const unsigned char* __restrict__ Mask, // [B,1000] (bool)
    const float* __restrict__ Wk1,          // [128,128]
    const float* __restrict__ Wv,           // [128,128]
    const float* __restrict__ Wk2,          // [128,128]
    const float* __restrict__ Wqf,          // [128,128]
    const float* __restrict__ Wout,         // [128,128]
    const float* __restrict__ Wqs,          // [130,128]
    float* __restrict__ Out)                // [B,1000]
{
    constexpr int N = N_NODES;
    constexpr int D = DIM;

    const int b    = blockIdx.x;
    const int tid  = threadIdx.x;
    const int lane = tid & 31;
    const int wave = tid >> 5;
    const int mrow = lane & 15;   // A-row / B-col / C-col owned by this lane
    const int ksel = lane >> 4;   // K sub-pair select (0: K,K+1 ; 1: K+2,K+3)

    const float* Xb = X + (size_t)b * N * D;
    const unsigned char* maskb = Mask + (size_t)b * N;

    __shared__ float tileL[2][128 * TSTRIDE]; // X tile ping-pong      (132 KB)
    __shared__ float sLds[8][NP];             // scores -> attn        ( 32 KB)
    __shared__ float uLds[16][D];             // folded Wk1*q, padded  (  8 KB)
    __shared__ float yLds[8][D];              // attn-weighted X sums  (  4 KB)
    __shared__ float gL[D];
    __shared__ float scL[130];
    __shared__ float q1L[D];
    __shared__ float mhaL[D];
    __shared__ float q2L[D];
    __shared__ float tL[D];

    // Prologue: start streaming tile 0 while we do the tiny projections.
    async_tile_load(Xb, tileL[0], 0, tid);

    // ---------- Phase 0: stage graph embedding + step context ----------
    if (tid < D)   gL[tid]  = Gemb[(size_t)b * D + tid];
    if (tid < 130) scL[tid] = Sctx[(size_t)b * 130 + tid];
    __syncthreads();

    // ---------- Phase 1: Q1 = g@Wq_fixed + sc@Wq_step ----------
    if (tid < D) {
        float acc = 0.f;
        for (int j = 0; j < D; ++j)   acc += gL[j]  * Wqf[j * D + tid];
        for (int j = 0; j < 130; ++j) acc += scL[j] * Wqs[j * D + tid];
        q1L[tid] = acc;
    }
    __syncthreads();

    // ---------- Phase 2: u[h][d] = (1/sqrt(16)) * sum_j Wk1[d][16h+j]*Q1[16h+j] ----------
    for (int e = tid; e < 16 * D; e += 256) {
        const int h = e >> 7;
        const int d = e & (D - 1);
        float val = 0.f;
        if (h < 8) {
            const float* wr = Wk1 + (size_t)d * D + h * 16;
            const float* qv = q1L + h * 16;
            float acc = 0.f;
#pragma unroll
            for (int j = 0; j < 16; ++j) acc += wr[j] * qv[j];
            val = 0.25f * acc;
        }
        (&uLds[0][0])[e] = val;   // heads 8..15 zero-padded for WMMA N=16
    }
    __syncthreads();

    // ---------- Phase 3: scores S(16 nodes x 16 heads) = Xtile @ u^T ----------
    // Double-buffered async-staged X tiles; WMMA A from LDS tile, B from uLds.
    for (int it = 0; it < 8; ++it) {
        const int cur = it & 1;
        wait_async0();          // my async issues landed
        __syncthreads();        // whole tile[cur] visible; prev buffer reads done
        if (it + 1 < 8) async_tile_load(Xb, tileL[cur ^ 1], (it + 1) * 128, tid);

        const float* arow = &tileL[cur][(wave * 16 + mrow) * TSTRIDE];
        const float* urow = uLds[mrow];
        v8f acc = {};
#pragma unroll
        for (int k = 0; k < D; k += 4) {
            const int kk = k + 2 * ksel;
            v2f a  = *(const v2f*)(arow + kk);   // A[row][kk,kk+1]
            v2f bm = *(const v2f*)(urow + kk);   // B[kk,kk+1][col=head]
            acc = __builtin_amdgcn_wmma_f32_16x16x4_f32(
                      false, a, false, bm, (short)0, acc, false, false);
        }
        const int ntile = it * 128 + wave * 16;
        if (mrow < 8) {                          // valid head columns only
#pragma unroll
            for (int r = 0; r < 8; ++r)
                sLds[mrow][ntile + 8 * ksel + r] = acc[r];
        }
    }
    __syncthreads();

    // ---------- Phase 4: masked softmax per head (wave w owns head w) ----------
    {
        const int h = wave;
        float m = -3.0e38f;
        for (int i = lane; i < NP; i += 32) {
            float v;
            if (i < N) v = maskb[i] ? NEG_INF : sLds[h][i];
            else       v = -1e30f;               // padding: below NEG_INF
            sLds[h][i] = v;
            m = fmaxf(m, v);
        }
        for (int off = 16; off; off >>= 1) m = fmaxf(m, __shfl_xor(m, off));
        float lsum = 0.f;
        for (int i = lane; i < NP; i += 32) {
            float p = __expf(sLds[h][i] - m);
            sLds[h][i] = p;
            lsum += p;
        }
        for (int off = 16; off; off >>= 1) lsum += __shfl_xor(lsum, off);
        const float inv = 1.0f / lsum;
        for (int i = lane; i < NP; i += 32) sLds[h][i] *= inv;
    }
    __syncthreads();

    // ---------- Phase 5: y(16 heads x 128) = attn @ X, async-staged tiles ----------
    async_tile_load(Xb, tileL[0], 0, tid);       // safe: all tile reads done above
    {
        const int col = wave * 16 + mrow;        // wave owns 16-col slice of y
        v8f acc = {};
        for (int ch = 0; ch < 8; ++ch) {
            const int cur = ch & 1;
            wait_async0();
            __syncthreads();
            if (ch + 1 < 8) async_tile_load(Xb, tileL[cur ^ 1], (ch + 1) * 128, tid);

            const float* tb = tileL[cur];
            const int kbase = ch * 128;
#pragma unroll 8
            for (int k = 0; k < 128; k += 4) {
                const int kk = k + 2 * ksel;
                const int hh = mrow & 7;
                float a0 = sLds[hh][kbase + kk];
                float a1 = sLds[hh][kbase + kk + 1];
                v2f a;
                a[0] = (mrow < 8) ? a0 : 0.f;    // pad head rows 8..15
                a[1] = (mrow < 8) ? a1 : 0.f;
                v2f bm;
                bm[0] = tb[kk * TSTRIDE + col];  // B[k][col] = X[node][col]
                bm[1] = tb[(kk + 1) * TSTRIDE + col];
                acc = __builtin_amdgcn_wmma_f32_16x16x4_f32(
                          false, a, false, bm, (short)0, acc, false, false);
            }
        }
        if (ksel == 0) {                         // head rows 0..7 live here
#pragma unroll
            for (int r = 0; r < 8; ++r) yLds[r][col] = acc[r];
        }
    }
    __syncthreads();

    // ---------- Phase 6: mha -> Q2 -> t (tiny per-batch VALU) ----------
    if (tid < D) {                               // mha[e] = sum_d y[h][d]*Wv[d][e]
        const int h = tid >> 4;
        float acc = 0.f;
        for (int d = 0; d < D; ++d) acc += yLds[h][d] * Wv[(size_t)d * D + tid];
        mhaL[tid] = acc;
    }
    __syncthreads();
    if (tid < D) {                               // Q2 = mha @ Wout
        float acc = 0.f;
        for (int e = 0; e < D; ++e) acc += mhaL[e] * Wout[(size_t)e * D + tid];
        q2L[tid] = acc;
    }
    __syncthreads();
    if (tid < D) {                               // t[d] = (1/sqrt(128)) * Wk2[d,:] . Q2
        const float* wr = Wk2 + (size_t)tid * D;
        float acc = 0.f;
        for (int c = 0; c < D; ++c) acc += wr[c] * q2L[c];
        tL[tid] = acc * 0.08838834764831845f;
    }
    __syncthreads();

    // ---------- Phase 7: logits[n] = mask ? -1e9 : 10*tanh(X[n,:] . t) ----------
    for (int n = tid; n < N; n += 256) {
        const float4* xr = (const float4*)(Xb + (size_t)n * D);
        float dot = 0.f;
#pragma unroll
        for (int j = 0; j < D / 4; ++j) {
            float4 xv = xr[j];
            dot += xv.x * tL[4*j] + xv.y * tL[4*j+1] + xv.z * tL[4*j+2] + xv.w * tL[4*j+3];
        }
        const float logit = maskb[n] ? NEG_INF : 10.0f * tanhf(dot);
        Out[(size_t)b * N + n] = logit;
    }
}

extern "C" void kernel_launch(void* const* d_in, const int* in_sizes, int n_in,
                              void* d_out, int out_size, void* d_ws, size_t ws_size,
                              hipStream_t stream) {
    const float* X            = (const float*)d_in[0];
    const float* G            = (const float*)d_in[1];
    const float* SC           = (const float*)d_in[2];
    const unsigned char* Mask = (const unsigned char*)d_in[3];
    const float* Wk1          = (const float*)d_in[4];
    const float* Wv           = (const float*)d_in[5];
    const float* Wk2          = (const float*)d_in[6];
    const float* Wqf          = (const float*)d_in[7];
    const float* Wout         = (const float*)d_in[8];
    const float* Wqs          = (const float*)d_in[9];
    float* Out                = (float*)d_out;

    decoder_cell_kernel<<<dim3(256), dim3(256), 0, stream>>>(
        X, G, SC, Mask, Wk1, Wv, Wk2, Wqf, Wout, Wqs, Out);
}